// BloomHashTripleFilter_32289564131584
// MI455X (gfx1250) — compile-verified
//
#include <hip/hip_runtime.h>
#include <stdint.h>

// ---------------------------------------------------------------------------
// BloomHashTripleFilter for MI455X (gfx1250, wave32).
//
// Reference semantics: out = ~(bloom_maybe & exact_found). Since every true
// hash sets its own bloom bits at build time, exact_found implies
// bloom_maybe, so out == ~exact_found. We therefore skip the bloom stage and
// implement exact membership with an open-addressing hash table that fits in
// the 192MB L2 (2^24 slots * 8B = 128MB, load factor ~0.30).
// ---------------------------------------------------------------------------

typedef int      v4i __attribute__((ext_vector_type(4)));
typedef float    v4f __attribute__((ext_vector_type(4)));
typedef unsigned v4u __attribute__((ext_vector_type(4)));

static constexpr int  B_E = 17;                       // ceil(log2(100001))
static constexpr int  B_R = 9;                        // ceil(log2(501))
static constexpr unsigned long long EMPTY_SLOT = ~0ull;      // hashes < 2^43
static constexpr unsigned long long HASH_MUL   = 0x9E3779B97F4A7C15ull;

__device__ __forceinline__ unsigned long long triple_hash(int h, int r, int t) {
    // Exactly mirrors reference: (h << (B_E+B_R)) | (r << B_E) | t
    return ((unsigned long long)(unsigned)h << (B_E + B_R)) |
           ((unsigned long long)(unsigned)r << B_E) |
           (unsigned long long)(unsigned)t;
}

__device__ __forceinline__ unsigned slot_of(unsigned long long h, int shift) {
    // Multiplicative mix -> top log2(slots) bits (hash fields are structured,
    // low table bits alone would cluster).
    return (unsigned)((h * HASH_MUL) >> shift);
}

// ---------------------------------------------------------------------------
// 1) Fill table with EMPTY sentinel (16B stores, full-bandwidth streaming).
// ---------------------------------------------------------------------------
__global__ void init_table_kernel(v4u* __restrict__ table, int n_vec) {
    int i = blockIdx.x * blockDim.x + threadIdx.x;
    if (i < n_vec) {
        v4u v = {~0u, ~0u, ~0u, ~0u};
        table[i] = v;
    }
}

// ---------------------------------------------------------------------------
// 2) Insert 5M true-triple hashes. 4 triples/thread: 3 x b128 NT loads
//    (48B contiguous per thread), then 4 independent linear-probe inserts.
// ---------------------------------------------------------------------------
__global__ void insert_kernel(const int* __restrict__ tt, long long n_triples,
                              unsigned long long* __restrict__ table,
                              unsigned slot_mask, int shift) {
    long long base = ((long long)blockIdx.x * blockDim.x + threadIdx.x) * 4;
    if (base >= n_triples) return;

    unsigned long long hv[4];
    int cnt;
    if (base + 4 <= n_triples) {
        const v4i* p = (const v4i*)(tt + base * 3);        // 16B aligned
        v4i a = __builtin_nontemporal_load(p + 0);
        v4i b = __builtin_nontemporal_load(p + 1);
        v4i c = __builtin_nontemporal_load(p + 2);
        hv[0] = triple_hash(a.x, a.y, a.z);
        hv[1] = triple_hash(a.w, b.x, b.y);
        hv[2] = triple_hash(b.z, b.w, c.x);
        hv[3] = triple_hash(c.y, c.z, c.w);
        cnt = 4;
    } else {
        cnt = (int)(n_triples - base);
        for (int k = 0; k < cnt; ++k) {
            const int* t = tt + (base + k) * 3;
            hv[k] = triple_hash(t[0], t[1], t[2]);
        }
    }

    for (int k = 0; k < cnt; ++k) {
        unsigned long long h = hv[k];
        unsigned idx = slot_of(h, shift);
        for (unsigned tries = 0; tries <= slot_mask; ++tries) {   // hang guard
            unsigned long long prev = atomicCAS(&table[idx], EMPTY_SLOT, h);
            if (prev == EMPTY_SLOT || prev == h) break;           // inserted / dup
            idx = (idx + 1) & slot_mask;
        }
    }
}

// ---------------------------------------------------------------------------
// 3) Query 8M triples. First probe of all 4 hashes issued as independent
//    loads (MLP, L2-resident); rare collision tail handled in a loop.
//    Output: 1.0f = keep (not a known true triple), 0.0f = filter out.
// ---------------------------------------------------------------------------
__device__ __forceinline__ float probe_rest(const unsigned long long* __restrict__ table,
                                            unsigned long long h, unsigned idx,
                                            unsigned slot_mask, unsigned long long v) {
    for (unsigned tries = 0; tries <= slot_mask; ++tries) {       // hang guard
        if (v == h)          return 0.0f;   // exact member -> filter out
        if (v == EMPTY_SLOT) return 1.0f;   // definitely absent -> keep
        idx = (idx + 1) & slot_mask;
        v = table[idx];
    }
    return 1.0f;
}

__global__ void query_kernel(const int* __restrict__ q, long long n_triples,
                             const unsigned long long* __restrict__ table,
                             unsigned slot_mask, int shift,
                             float* __restrict__ out) {
    long long base = ((long long)blockIdx.x * blockDim.x + threadIdx.x) * 4;
    if (base >= n_triples) return;

    if (base + 4 <= n_triples) {
        const v4i* p = (const v4i*)(q + base * 3);         // 16B aligned
        v4i a = __builtin_nontemporal_load(p + 0);
        v4i b = __builtin_nontemporal_load(p + 1);
        v4i c = __builtin_nontemporal_load(p + 2);

        unsigned long long h0 = triple_hash(a.x, a.y, a.z);
        unsigned long long h1 = triple_hash(a.w, b.x, b.y);
        unsigned long long h2 = triple_hash(b.z, b.w, c.x);
        unsigned long long h3 = triple_hash(c.y, c.z, c.w);

        unsigned i0 = slot_of(h0, shift);
        unsigned i1 = slot_of(h1, shift);
        unsigned i2 = slot_of(h2, shift);
        unsigned i3 = slot_of(h3, shift);

        // Four independent first-probe loads -> overlapped L2 latency.
        unsigned long long v0 = table[i0];
        unsigned long long v1 = table[i1];
        unsigned long long v2 = table[i2];
        unsigned long long v3 = table[i3];

        v4f r;
        r.x = probe_rest(table, h0, i0, slot_mask, v0);
        r.y = probe_rest(table, h1, i1, slot_mask, v1);
        r.z = probe_rest(table, h2, i2, slot_mask, v2);
        r.w = probe_rest(table, h3, i3, slot_mask, v3);
        __builtin_nontemporal_store(r, (v4f*)(out + base));
    } else {
        for (long long i = base; i < n_triples; ++i) {
            const int* t = q + i * 3;
            unsigned long long h = triple_hash(t[0], t[1], t[2]);
            unsigned idx = slot_of(h, shift);
            out[i] = probe_rest(table, h, idx, slot_mask, table[idx]);
        }
    }
}

// ---------------------------------------------------------------------------
// Host launcher: init table -> insert true hashes -> probe queries.
// Table lives in d_ws; sized to the largest power of two that fits
// (target 2^24 slots = 128MB, inside the 192MB L2).
// ---------------------------------------------------------------------------
extern "C" void kernel_launch(void* const* d_in, const int* in_sizes, int n_in,
                              void* d_out, int out_size, void* d_ws, size_t ws_size,
                              hipStream_t stream) {
    (void)n_in; (void)out_size;

    const int* true_triples = (const int*)d_in[0];   // (N_TRUE, 3) ints
    const int* triples      = (const int*)d_in[1];   // (N_QUERY, 3) ints
    float*     out          = (float*)d_out;         // (N_QUERY,) 1.0=keep

    long long n_true  = (long long)in_sizes[0] / 3;
    long long n_query = (long long)in_sizes[1] / 3;

    int log2_slots = 24;                             // 128MB preferred
    while (log2_slots > 20 && (8ull << log2_slots) > ws_size) --log2_slots;
    unsigned slots     = 1u << log2_slots;
    unsigned slot_mask = slots - 1u;
    int      shift     = 64 - log2_slots;
    unsigned long long* table = (unsigned long long*)d_ws;

    const int BLOCK = 256;                           // 8 wave32 waves / block

    {   // init: slots/2 x 16B stores
        int n_vec   = (int)(slots / 2);
        int nblocks = (n_vec + BLOCK - 1) / BLOCK;
        init_table_kernel<<<nblocks, BLOCK, 0, stream>>>((v4u*)d_ws, n_vec);
    }
    {   // build
        long long threads = (n_true + 3) / 4;
        int nblocks = (int)((threads + BLOCK - 1) / BLOCK);
        insert_kernel<<<nblocks, BLOCK, 0, stream>>>(true_triples, n_true,
                                                     table, slot_mask, shift);
    }
    {   // query
        long long threads = (n_query + 3) / 4;
        int nblocks = (int)((threads + BLOCK - 1) / BLOCK);
        query_kernel<<<nblocks, BLOCK, 0, stream>>>(triples, n_query,
                                                    table, slot_mask, shift, out);
    }
}